// CombinedPriorityLoss_609885356367
// MI455X (gfx1250) — compile-verified
//
#include <hip/hip_runtime.h>

typedef float v2f __attribute__((ext_vector_type(2)));
typedef float v8f __attribute__((ext_vector_type(8)));

#define MARGIN   0.1f
#define MSE_W    0.4f
#define RANK_W   0.6f
#define CHUNK    2048            // column elements staged in LDS per sweep (8 KB x2)

#if defined(__HIP_DEVICE_COMPILE__)
#if !__has_builtin(__builtin_amdgcn_wmma_f32_16x16x4_f32)
#error "wmma_f32_16x16x4_f32 builtin not available on this toolchain (device pass)"
#endif
#endif

// ---------------------------------------------------------------------------
// Zero the workspace accumulators (rank hinge sum + pair count).
// ---------------------------------------------------------------------------
__global__ void cpl_init_ws(float* rank_sum, unsigned int* cnt) {
  *rank_sum = 0.0f;
  *cnt = 0u;
}

// ---------------------------------------------------------------------------
// Pairwise ranking kernel.
//   - Block = 8 waves; each wave owns one 16-row strip of the NxN pair matrix.
//   - Columns are staged cooperatively into LDS in 2048-element chunks
//     (sentinels written at staging time: t=+inf, p=0 for the tail), so the
//     compute loop is branch-free and EXEC stays all-ones for WMMA.
//   - Per 16x16 tile, two V_WMMA_F32_16X16X4_F32 build the rank-2 matrices
//       Dif[i][j]  = p_j - p_i + margin   (A = [-p_i, 1], B = [1, p_j+margin])
//       Tdif[i][j] = t_i - t_j            (A = [ t_i, 1], B = [1, -t_j])
//     in the native C/D layout; VALU applies mask (Tdif>0), relu and
//     accumulates hinge sum (f32) + exact pair count (u32).
// ---------------------------------------------------------------------------
__global__ __launch_bounds__(256) void cpl_pair_rank(const float* __restrict__ p,
                                                     const float* __restrict__ t,
                                                     int N,
                                                     float* __restrict__ rank_sum,
                                                     unsigned int* __restrict__ cnt_out) {
  __shared__ float lds_p[CHUNK];
  __shared__ float lds_t[CHUNK];
  __shared__ float        ssum[256];
  __shared__ unsigned int scnt[256];

  const int lane        = threadIdx.x & 31;
  const int waveInBlock = threadIdx.x >> 5;
  const int waveId      = blockIdx.x * 8 + waveInBlock;   // one 16-row tile per wave
  const int numTiles    = (N + 15) >> 4;
  const int numChunks   = (N + CHUNK - 1) / CHUNK;

  const bool lo  = (lane < 16);
  const int  l15 = lane & 15;

  // --- row (i) fragments, invariant over the whole sweep (branchless) -------
  const int  i      = waveId * 16 + l15;
  const bool ivalid = (i < N);                 // false also for "extra" waves
  const int  ic     = ivalid ? i : 0;
  const float pi    = ivalid ? p[ic] : 0.0f;
  const float ti    = ivalid ? t[ic] : -__builtin_inff();  // -inf row: mask never passes

  // A is 16x4 f32: lanes0-15 -> {K0,K1}, lanes16-31 -> {K2,K3} (zeroed)
  v2f aD, aT;
  aD.x = lo ? -pi : 0.0f;   aD.y = lo ? 1.0f : 0.0f;
  aT.x = lo ?  ti : 0.0f;   aT.y = aD.y;

  float        sum = 0.0f;
  unsigned int cnt = 0u;

  for (int chunk = 0; chunk < numChunks; ++chunk) {
    const int chunkBase = chunk * CHUNK;

    // --- cooperative staging with sentinel padding (whole block) ------------
    for (int e = threadIdx.x; e < CHUNK; e += 256) {
      const int  g  = chunkBase + e;
      const bool v  = (g < N);
      const int  gc = v ? g : 0;
      const float pv = p[gc];
      const float tv = t[gc];
      lds_p[e] = v ? pv : 0.0f;
      lds_t[e] = v ? tv : __builtin_inff();    // +inf col: mask never passes
    }
    __syncthreads();

    const int tilesHere = min(CHUNK / 16, numTiles - chunk * (CHUNK / 16));

#pragma unroll 2
    for (int tt = 0; tt < tilesHere; ++tt) {
      const float pj = lds_p[tt * 16 + l15];
      const float tj = lds_t[tt * 16 + l15];

      // B is 4x16 f32: lanes0-15 -> rows {K0,K1}, lanes16-31 -> {K2,K3} (zeroed)
      v2f bD, bT;
      bD.x = lo ? 1.0f : 0.0f;   bD.y = lo ? (pj + MARGIN) : 0.0f;
      bT.x = bD.x;               bT.y = lo ? -tj : 0.0f;

#if defined(__HIP_DEVICE_COMPILE__)
      v8f z = {};
      v8f dif = __builtin_amdgcn_wmma_f32_16x16x4_f32(
          /*neg_a=*/false, aD, /*neg_b=*/false, bD, /*c_mod=*/(short)0, z,
          /*reuse_a=*/false, /*reuse_b=*/false);
      v8f tdf = __builtin_amdgcn_wmma_f32_16x16x4_f32(
          /*neg_a=*/false, aT, /*neg_b=*/false, bT, /*c_mod=*/(short)0, z,
          /*reuse_a=*/false, /*reuse_b=*/false);
#else
      v8f dif = {}, tdf = {};
      (void)bD; (void)bT;
#endif

#pragma unroll
      for (int r = 0; r < 8; ++r) {
        const bool m = (tdf[r] > 0.0f);        // t_i > t_j (exact via diff sign)
        float h = dif[r];
        h = (h > 0.0f) ? h : 0.0f;             // relu(p_j - p_i + margin)
        sum += m ? h : 0.0f;
        cnt += m ? 1u : 0u;
      }
    }
    __syncthreads();
  }

  // --- block reduction (LDS tree), then one atomic per block ----------------
  ssum[threadIdx.x] = sum;
  scnt[threadIdx.x] = cnt;
  __syncthreads();
  for (int s = 128; s > 0; s >>= 1) {
    if ((int)threadIdx.x < s) {
      ssum[threadIdx.x] += ssum[threadIdx.x + s];
      scnt[threadIdx.x] += scnt[threadIdx.x + s];
    }
    __syncthreads();
  }
  if (threadIdx.x == 0) {
    atomicAdd(rank_sum, ssum[0]);
    atomicAdd(cnt_out, scnt[0]);
  }
}

// ---------------------------------------------------------------------------
// MSE + final combine (single block; stream-ordered after the pair kernel).
// ---------------------------------------------------------------------------
__global__ __launch_bounds__(256) void cpl_mse_finalize(const float* __restrict__ p,
                                                        const float* __restrict__ t,
                                                        int N,
                                                        const float* __restrict__ rank_sum,
                                                        const unsigned int* __restrict__ cnt,
                                                        float* __restrict__ out) {
  __shared__ float s[256];
  float acc = 0.0f;
  for (int i = threadIdx.x; i < N; i += 256) {
    float d = p[i] - t[i];
    acc += d * d;
  }
  s[threadIdx.x] = acc;
  __syncthreads();
  for (int st = 128; st > 0; st >>= 1) {
    if ((int)threadIdx.x < st) s[threadIdx.x] += s[threadIdx.x + st];
    __syncthreads();
  }
  if (threadIdx.x == 0) {
    float mse = s[0] / (float)N;
    unsigned int c = *cnt;
    float denom = (c > 0u) ? (float)c : 1.0f;
    float rank = *rank_sum / denom;
    out[0] = MSE_W * mse + RANK_W * rank;
  }
}

// ---------------------------------------------------------------------------
extern "C" void kernel_launch(void* const* d_in, const int* in_sizes, int n_in,
                              void* d_out, int out_size, void* d_ws, size_t ws_size,
                              hipStream_t stream) {
  const float* p = (const float*)d_in[0];   // predictions
  const float* t = (const float*)d_in[1];   // targets
  const int N = in_sizes[0];

  float*        rank_sum = (float*)d_ws;                 // ws[0]: f32 hinge sum
  unsigned int* cnt      = ((unsigned int*)d_ws) + 1;    // ws[1]: u32 pair count

  cpl_init_ws<<<1, 1, 0, stream>>>(rank_sum, cnt);

  const int numTiles = (N + 15) / 16;        // 512 row tiles for N=8192
  const int blocks   = (numTiles + 7) / 8;   // 8 waves (one tile each) per block
  cpl_pair_rank<<<blocks, 256, 0, stream>>>(p, t, N, rank_sum, cnt);

  cpl_mse_finalize<<<1, 256, 0, stream>>>(p, t, N, rank_sum, cnt, (float*)d_out);
}